// GCN_67018669686888
// MI455X (gfx1250) — compile-verified
//
#include <hip/hip_runtime.h>
#include <hip/hip_bf16.h>

typedef __attribute__((ext_vector_type(2))) float v2f;
typedef __attribute__((ext_vector_type(8))) float v8f;

#define FD 128  // feature dim

// ---- degree / norm prep -------------------------------------------------
__global__ void k_fill1(float* __restrict__ p, int n) {
  int i = blockIdx.x * blockDim.x + threadIdx.x;
  if (i < n) p[i] = 1.0f;  // self-loop contributes 1 to every degree
}

__global__ void k_degcount(const int* __restrict__ dst, float* __restrict__ deg, int e) {
  int i = blockIdx.x * blockDim.x + threadIdx.x;
  if (i < e) atomicAdd(&deg[dst[i]], 1.0f);
}

__global__ void k_rsqrt(float* __restrict__ p, int n) {
  int i = blockIdx.x * blockDim.x + threadIdx.x;
  if (i < n) p[i] = rsqrtf(p[i]);
}

// ---- seed output with bias (clears harness poison); float4 stores ------
// nvec = N*D/4; each thread writes one float4. Bias index: (i*4) & 127
// -> float4 of bias at (i & 31). 128 % 4 == 0 so vectors never straddle.
__global__ void k_bias_init(float4* __restrict__ out, const float4* __restrict__ b4,
                            long long nvec) {
  long long i = (long long)blockIdx.x * blockDim.x + threadIdx.x;
  if (i < nvec) out[i] = b4[i & (FD / 4 - 1)];
}

// ---- GEMM: T = (relu? max(A,0) : A) @ W, fp32 WMMA 16x16x4 --------------
// One wave computes a 16-row x 128-col strip: 8 C tiles (8 VGPRs each),
// K swept in steps of 4 -> 8 v_wmma_f32_16x16x4_f32 per step.
// A layout (32-bit A 16x4): lanes 0-15: M=l, v0=K, v1=K+1 ; lanes 16-31: K+2,K+3.
// B layout (4x16):          lanes 0-15: N=l, v0=K, v1=K+1 ; lanes 16-31: K+2,K+3.
__global__ void __launch_bounds__(128)
k_gemm_wmma(const float* __restrict__ A, const float* __restrict__ W,
            float* __restrict__ T, int strips, int relu) {
  int wave = blockIdx.x * (blockDim.x >> 5) + (threadIdx.x >> 5);
  if (wave >= strips) return;               // wave-uniform: EXEC stays all-1s
  int lane = threadIdx.x & 31;
  int half = lane >> 4;                     // 0: K pair (k,k+1); 1: (k+2,k+3)
  int l    = lane & 15;

  const float* arow = A + (size_t)(wave * 16 + l) * FD + 2 * half;
  v8f c[8] = {};

  for (int k0 = 0; k0 < FD; k0 += 4) {
    v2f a = *(const v2f*)(arow + k0);       // b64 load: (K, K+1) contiguous
    if (relu) {
      a.x = fmaxf(a.x, 0.0f);
      a.y = fmaxf(a.y, 0.0f);
    }
    const float* w0 = W + (size_t)(k0 + 2 * half) * FD + l;
#pragma unroll
    for (int ct = 0; ct < 8; ++ct) {
      v2f b; b.x = w0[ct * 16]; b.y = w0[ct * 16 + FD];
      c[ct] = __builtin_amdgcn_wmma_f32_16x16x4_f32(
          /*neg_a=*/false, a, /*neg_b=*/false, b,
          /*c_mod=*/(short)0, c[ct], /*reuse_a=*/false, /*reuse_b=*/false);
    }
  }

  // C/D layout: VGPR v -> row = 16*wave + v + 8*half, col = ct*16 + l
#pragma unroll
  for (int ct = 0; ct < 8; ++ct) {
    float* o = T + (size_t)(wave * 16 + 8 * half) * FD + ct * 16 + l;
#pragma unroll
    for (int v = 0; v < 8; ++v) o[(size_t)v * FD] = c[ct][v];
  }
}

// ---- edge aggregation: out[dst] += norm * T[src] (L2-resident atomics) --
// One wave per edge (idx < e) or self-loop (idx >= e). Each lane handles
// one float4 (128 floats / 32 lanes): b128 load + 4 global_atomic_add_f32.
__global__ void __launch_bounds__(256)
k_aggregate(const float* __restrict__ t, const int* __restrict__ src,
            const int* __restrict__ dst, const float* __restrict__ dinv,
            float* __restrict__ out, int e, int n) {
  int idx  = blockIdx.x * (blockDim.x >> 5) + (threadIdx.x >> 5);
  int lane = threadIdx.x & 31;
  if (idx >= e + n) return;
  int s, d; float nrm;
  if (idx < e) {
    s = src[idx]; d = dst[idx];
    nrm = dinv[s] * dinv[d];
  } else {
    s = idx - e; d = s;
    float r = dinv[s]; nrm = r * r;
  }
  const float4* tp = (const float4*)(t + (size_t)s * FD);
  float4 v = tp[lane];
  float* op = out + (size_t)d * FD + lane * 4;
  atomicAdd(op + 0, v.x * nrm);
  atomicAdd(op + 1, v.y * nrm);
  atomicAdd(op + 2, v.z * nrm);
  atomicAdd(op + 3, v.w * nrm);
}

extern "C" void kernel_launch(void* const* d_in, const int* in_sizes, int n_in,
                              void* d_out, int out_size, void* d_ws, size_t ws_size,
                              hipStream_t stream) {
  const float* x  = (const float*)d_in[0];
  const int*   ei = (const int*)d_in[1];
  const float* W1 = (const float*)d_in[2];
  const float* b1 = (const float*)d_in[3];
  const float* W2 = (const float*)d_in[4];
  const float* b2 = (const float*)d_in[5];
  const float* W3 = (const float*)d_in[6];
  const float* b3 = (const float*)d_in[7];

  const int Nn = in_sizes[0] / FD;   // 100000 (multiple of 16)
  const int Ee = in_sizes[1] / 2;    // 1600000
  const int* srcp = ei;
  const int* dstp = ei + Ee;

  float* dinv = (float*)d_ws;                 // N floats
  float* t    = dinv + Nn;                    // N*D floats (GEMM out)
  float* h    = t + (size_t)Nn * FD;          // N*D floats (layer out)
  float* out  = (float*)d_out;

  const int strips = Nn / 16;
  dim3 gemmBlk(128);
  dim3 gemmGrd((strips + 3) / 4);
  const long long ndv = (long long)Nn * FD / 4;   // float4 count
  int grdNDv = (int)((ndv + 255) / 256);
  int grdAgg = (Ee + Nn + 7) / 8;            // 8 waves (edges) per 256-thr block
  int grdN   = (Nn + 255) / 256;
  int grdE   = (Ee + 255) / 256;

  // degree -> dinv = rsqrt(deg), deg includes self-loop
  k_fill1   <<<grdN, 256, 0, stream>>>(dinv, Nn);
  k_degcount<<<grdE, 256, 0, stream>>>(dstp, dinv, Ee);
  k_rsqrt   <<<grdN, 256, 0, stream>>>(dinv, Nn);

  // layer 1: t = x@W1 ; h = agg(t) + b1        (relu folded into next gemm)
  k_gemm_wmma<<<gemmGrd, gemmBlk, 0, stream>>>(x, W1, t, strips, 0);
  k_bias_init<<<grdNDv, 256, 0, stream>>>((float4*)h, (const float4*)b1, ndv);
  k_aggregate<<<grdAgg, 256, 0, stream>>>(t, srcp, dstp, dinv, h, Ee, Nn);

  // layer 2: t = relu(h)@W2 ; h = agg(t) + b2
  k_gemm_wmma<<<gemmGrd, gemmBlk, 0, stream>>>(h, W2, t, strips, 1);
  k_bias_init<<<grdNDv, 256, 0, stream>>>((float4*)h, (const float4*)b2, ndv);
  k_aggregate<<<grdAgg, 256, 0, stream>>>(t, srcp, dstp, dinv, h, Ee, Nn);

  // layer 3: t = relu(h)@W3 ; out = agg(t) + b3
  k_gemm_wmma<<<gemmGrd, gemmBlk, 0, stream>>>(h, W3, t, strips, 1);
  k_bias_init<<<grdNDv, 256, 0, stream>>>((float4*)out, (const float4*)b3, ndv);
  k_aggregate<<<grdAgg, 256, 0, stream>>>(t, srcp, dstp, dinv, out, Ee, Nn);
}